// DictionaryLearning_90890097918487
// MI455X (gfx1250) — compile-verified
//
#include <hip/hip_runtime.h>

// ---------------------------------------------------------------------------
// DictionaryLearning forward (batch OMP) for gfx1250 (MI455X, wave32).
//   Y : [16384, 256] f32, D : [512, 256] f32 (rows unit-norm)
//   out = concat(Y_pred [16384,256], X [16384,512])
// Pipeline:
//   1) gram_wmma : G = D D^T + eps*I           (v_wmma_f32_16x16x4_f32)
//   2) hbar_wmma : h_bar = Y D^T, 16x64/wave   (4 independent wmma chains)
//   3) omp_kernel: per-signal OMP, one wave32 per signal; sparse Y_pred.
// ---------------------------------------------------------------------------

typedef __attribute__((ext_vector_type(2))) float v2f;
typedef __attribute__((ext_vector_type(8))) float v8f;

#define Bsz   16384
#define Mdim  256
#define Ndim  512
#define KMAX  8
#define DIAG_EPS 1e-5f

// WMMA fragment mapping assumed (ISA 7.12.2, 32-bit operands, wave32):
//   A (16x4)  lane L: M = L&15, holds K = v + 2*(L>>4) in VGPR v   -> float2
//   B (4x16)  lane L: N = L&15, holds K = v + 2*(L>>4) in VGPR v   -> float2
//   C/D(16x16)lane L: N = L&15, M = v + 8*(L>>4) in VGPR v
// Both A and B fragments are contiguous float2 loads from row-major storage.

static __device__ __forceinline__ v8f wmma_f32(v2f a, v2f b, v8f c) {
  return __builtin_amdgcn_wmma_f32_16x16x4_f32(false, a, false, b,
                                               (short)0, c, false, false);
}

// -------------------- G = D D^T + eps I  (512x512, K=256) -------------------
__global__ __launch_bounds__(32) void gram_wmma(const float* __restrict__ D,
                                                float* __restrict__ G) {
  const int i0 = blockIdx.x * 16;   // output rows
  const int j0 = blockIdx.y * 16;   // output cols
  const int lane = threadIdx.x;
  const int half = lane >> 4;
  const int l15  = lane & 15;
  const float* arow = D + (size_t)(i0 + l15) * Mdim + 2 * half;
  const float* brow = D + (size_t)(j0 + l15) * Mdim + 2 * half;
  v8f c = {};
  for (int k0 = 0; k0 < Mdim; k0 += 4) {
    v2f a = *(const v2f*)(arow + k0);
    v2f b = *(const v2f*)(brow + k0);
    c = wmma_f32(a, b, c);
  }
#pragma unroll
  for (int v = 0; v < 8; ++v) {
    int row = i0 + v + 8 * half;
    int col = j0 + l15;
    float val = c[v];
    if (row == col) val += DIAG_EPS;
    G[(size_t)row * Ndim + col] = val;
  }
}

// ------------- h_bar = Y D^T  (16384x512, K=256), 16x64 per wave ------------
// 4 independent accumulators break the WMMA RAW chain; A fragment is reused
// across the 4 N-tiles (1 A load + 4 B loads per 4 WMMAs).
__global__ __launch_bounds__(128) void hbar_wmma(const float* __restrict__ Y,
                                                 const float* __restrict__ D,
                                                 float* __restrict__ hbar) {
  const int wave = threadIdx.x >> 5;
  const int lane = threadIdx.x & 31;
  const int b0 = (blockIdx.x * 4 + wave) * 16;   // signal rows (16)
  const int n0 = blockIdx.y * 64;                // atom cols  (64)
  const int half = lane >> 4;
  const int l15  = lane & 15;
  const float* yrow = Y + (size_t)(b0 + l15) * Mdim + 2 * half;
  const float* d0 = D + (size_t)(n0 + l15) * Mdim + 2 * half;
  const float* d1 = d0 + 16 * Mdim;
  const float* d2 = d0 + 32 * Mdim;
  const float* d3 = d0 + 48 * Mdim;
  v8f c0 = {}, c1 = {}, c2 = {}, c3 = {};
#pragma unroll 4
  for (int k0 = 0; k0 < Mdim; k0 += 4) {
    v2f a  = *(const v2f*)(yrow + k0);
    v2f b0v = *(const v2f*)(d0 + k0);
    v2f b1v = *(const v2f*)(d1 + k0);
    v2f b2v = *(const v2f*)(d2 + k0);
    v2f b3v = *(const v2f*)(d3 + k0);
    c0 = wmma_f32(a, b0v, c0);
    c1 = wmma_f32(a, b1v, c1);
    c2 = wmma_f32(a, b2v, c2);
    c3 = wmma_f32(a, b3v, c3);
  }
  float* outbase = hbar + (size_t)b0 * Ndim + n0 + l15;
#pragma unroll
  for (int v = 0; v < 8; ++v) {
    size_t roff = (size_t)(v + 8 * half) * Ndim;
    outbase[roff +  0] = c0[v];
    outbase[roff + 16] = c1[v];
    outbase[roff + 32] = c2[v];
    outbase[roff + 48] = c3[v];
  }
}

// -------------------- per-signal OMP: one wave32 per batch row --------------
// Lane l owns columns n = l + 32*t, t in [0,16). Small (<=8x8) linear algebra
// is replicated uniformly across all lanes (no divergence, no LDS).
__global__ __launch_bounds__(256) void omp_kernel(const float* __restrict__ hbar_g,
                                                  const float* __restrict__ G,
                                                  const float* __restrict__ D,
                                                  float* __restrict__ Ypred,
                                                  float* __restrict__ Xout) {
  const int wave = threadIdx.x >> 5;
  const int lane = threadIdx.x & 31;
  const int b = blockIdx.x * 8 + wave;

  const float* hb = hbar_g + (size_t)b * Ndim;
  float hbar_l[16], h_l[16];
#pragma unroll
  for (int t = 0; t < 16; ++t) {
    hbar_l[t] = hb[lane + 32 * t];
    h_l[t] = hbar_l[t];
  }

  unsigned selmask = 0;          // bit t: column lane+32t selected
  float Lm[36];                  // packed lower-tri: Lm[i*(i+1)/2 + j]
  int   Iv[8];
  float hIv[8];
  float xs[8];
  Lm[0] = 1.0f;                  // reference inits L = [[1]]

  for (int kk = 0; kk < KMAX; ++kk) {
    // ---- masked argmax of |h| (first index on ties, like jnp.argmax) ----
    float bestv = -1.0f;
    int   bestn = Ndim;
#pragma unroll
    for (int t = 0; t < 16; ++t) {
      float cv = ((selmask >> t) & 1u) ? -1.0f : fabsf(h_l[t]);
      int n = lane + 32 * t;
      if (cv > bestv) { bestv = cv; bestn = n; }
    }
#pragma unroll
    for (int off = 16; off >= 1; off >>= 1) {
      float ov = __shfl_xor(bestv, off, 32);
      int   on = __shfl_xor(bestn, off, 32);
      if (ov > bestv || (ov == bestv && on < bestn)) { bestv = ov; bestn = on; }
    }
    const int index = bestn;     // uniform across wave
    if ((index & 31) == lane) selmask |= 1u << (index >> 5);

    // ---- Cholesky row append: w = L^-1 g, corner = sqrt(diag - |w|^2) ----
    if (kk > 0) {
      float wv[8];
      float ss = 0.0f;
      for (int i = 0; i < kk; ++i) {
        float s = G[(size_t)Iv[i] * Ndim + index];
        for (int j = 0; j < i; ++j) s -= Lm[i * (i + 1) / 2 + j] * wv[j];
        wv[i] = s / Lm[i * (i + 1) / 2 + i];
        ss += wv[i] * wv[i];
      }
      float diag_g = G[(size_t)index * Ndim + index];
      float corner = sqrtf(fmaxf(diag_g - ss, DIAG_EPS));
      int base = kk * (kk + 1) / 2;
      for (int j = 0; j < kk; ++j) Lm[base + j] = wv[j];
      Lm[base + kk] = corner;
    }
    Iv[kk]  = index;
    hIv[kk] = hb[index];

    // ---- cholesky_solve: L y = h_I ; L^T x = y ----
    float yv[8];
    for (int i = 0; i <= kk; ++i) {
      float s = hIv[i];
      for (int j = 0; j < i; ++j) s -= Lm[i * (i + 1) / 2 + j] * yv[j];
      yv[i] = s / Lm[i * (i + 1) / 2 + i];
    }
    for (int i = kk; i >= 0; --i) {
      float s = yv[i];
      for (int j = i + 1; j <= kk; ++j) s -= Lm[j * (j + 1) / 2 + i] * xs[j];
      xs[i] = s / Lm[i * (i + 1) / 2 + i];
    }

    // ---- h = h_bar - x^T G[I,:]  (G rows L2-resident, coalesced 128B) ----
#pragma unroll
    for (int t = 0; t < 16; ++t) h_l[t] = hbar_l[t];
    for (int j = 0; j <= kk; ++j) {
      const float* grow = G + (size_t)Iv[j] * Ndim + lane;
      float xj = xs[j];
#pragma unroll
      for (int t = 0; t < 16; ++t) h_l[t] -= xj * grow[32 * t];
    }
  }

  // ---- scatter sparse codes into dense X ----
  float xl[16];
#pragma unroll
  for (int t = 0; t < 16; ++t) xl[t] = 0.0f;
#pragma unroll
  for (int j = 0; j < KMAX; ++j)
    if ((Iv[j] & 31) == lane) xl[Iv[j] >> 5] = xs[j];
  float* xo = Xout + (size_t)b * Ndim + lane;
#pragma unroll
  for (int t = 0; t < 16; ++t) xo[32 * t] = xl[t];

  // ---- Y_pred = X @ D exploiting 8-sparsity of X ----
  float* yo = Ypred + (size_t)b * Mdim + lane;
#pragma unroll
  for (int t = 0; t < 8; ++t) {
    int m = lane + 32 * t;
    float acc = 0.0f;
#pragma unroll
    for (int j = 0; j < KMAX; ++j)
      acc += xs[j] * D[(size_t)Iv[j] * Mdim + m];
    yo[32 * t] = acc;
  }
}

// ---------------------------------------------------------------------------
extern "C" void kernel_launch(void* const* d_in, const int* in_sizes, int n_in,
                              void* d_out, int out_size, void* d_ws, size_t ws_size,
                              hipStream_t stream) {
  (void)in_sizes; (void)n_in; (void)out_size; (void)ws_size;
  const float* Y = (const float*)d_in[0];   // [16384, 256]
  const float* D = (const float*)d_in[1];   // [512, 256]

  float* out   = (float*)d_out;
  float* Ypred = out;                                   // [16384, 256]
  float* Xout  = out + (size_t)Bsz * Mdim;              // [16384, 512]

  float* G    = (float*)d_ws;                           // [512, 512]   (1 MB)
  float* hbar = G + (size_t)Ndim * Ndim;                // [16384, 512] (32 MB)

  {
    dim3 grid(Ndim / 16, Ndim / 16);
    gram_wmma<<<grid, 32, 0, stream>>>(D, G);
  }
  {
    dim3 grid(Bsz / 64, Ndim / 64);
    hbar_wmma<<<grid, 128, 0, stream>>>(Y, D, hbar);
  }
  omp_kernel<<<Bsz / 8, 256, 0, stream>>>(hbar, G, D, Ypred, Xout);
}